// TransformerConv_28063316312097
// MI455X (gfx1250) — compile-verified
//
#include <hip/hip_runtime.h>
#include <math.h>

#define NNODES 100000
#define NEDGES 1600000
#define INF_ 64
#define HD_ 64

typedef float v2f __attribute__((ext_vector_type(2)));
typedef float v8f __attribute__((ext_vector_type(8)));
typedef unsigned int u32x4 __attribute__((ext_vector_type(4)));
typedef int i32x4 __attribute__((ext_vector_type(4)));
typedef int i32x8 __attribute__((ext_vector_type(8)));

// ---------------------------------------------------------------------------
// Kernel 1: fused node projections  q/k/v/skip = feat @ W{q,k,v,s} + b
// One block = 16 node rows, 4 waves; wave w computes the full 16x64 output for
// weight matrix w using V_WMMA_F32_16X16X4_F32 (64 WMMAs per wave).
// The 16x64 f32 A tile is staged into LDS by the Tensor Data Mover with
// hardware row padding (stride 68 floats -> bank = (4m+kk)%64, conflict-free).
// ---------------------------------------------------------------------------
__global__ __launch_bounds__(128) void proj_wmma_kernel(
    const float* __restrict__ feat,
    const float* __restrict__ Wq, const float* __restrict__ bq,
    const float* __restrict__ Wk, const float* __restrict__ bk,
    const float* __restrict__ Wv, const float* __restrict__ bv,
    const float* __restrict__ Wsk, const float* __restrict__ bsk,
    float* __restrict__ q, float* __restrict__ k,
    float* __restrict__ v, float* __restrict__ sk)
{
    __shared__ float At[16 * 68];

    const int t    = threadIdx.x;          // 0..127
    const int row0 = blockIdx.x * 16;      // N = 6250*16 exactly

    // --- TDM: DMA the 16x64 f32 tile feat[row0:row0+16, :] into LDS --------
    // D# group0: count=1, lds_addr, global_addr(57b), type=2 ("image").
    // D# group1: data_size=4B(code 2), pad_enable, pad_interval=5 (every 64
    //            DWORDs), pad_amount=3 (+4 DWORDs) => LDS row stride 68 floats;
    //            tensor_dim0=64, tensor_dim1=N, tile_dim0=64, tile_dim1=16,
    //            tensor_dim0_stride=64, tensor_dim1_stride=64.
    if (t < 32) {                           // one wave issues the DMA
        const unsigned long long ga =
            (unsigned long long)(uintptr_t)(feat + (long)row0 * INF_);
        const unsigned lds = (unsigned)(uintptr_t)At;   // LDS byte offset
        u32x4 g0 = { 1u,                                  // count=1
                     lds,                                 // bits 63:32
                     (unsigned)ga,                        // bits 95:64
                     ((unsigned)(ga >> 32) & 0x01FFFFFFu) | 0x80000000u }; // type=2
        i32x8 g1 = { (int)0x07520000u,                    // dsz=4B, pad 64->+4 dw
                     (int)(64u << 16),                    // tensor_dim0=64 (lo16)
                     (int)(0x86A0u << 16),                // tensor_dim1=100000 lo
                     (int)(0x1u | (64u << 16)),           // dim1 hi | tile_dim0=64
                     16,                                  // tile_dim1=16
                     64,                                  // tensor_dim0_stride=64
                     (int)(64u << 16),                    // tensor_dim1_stride=64
                     0 };
        i32x4 gz = { 0, 0, 0, 0 };
#if __clang_major__ >= 23
        i32x8 gz8 = { 0, 0, 0, 0, 0, 0, 0, 0 };
        __builtin_amdgcn_tensor_load_to_lds(g0, g1, gz, gz, gz8, 0);
#else
        __builtin_amdgcn_tensor_load_to_lds(g0, g1, gz, gz, 0);
#endif
        __builtin_amdgcn_s_wait_tensorcnt(0);
    }
    __syncthreads();

    const int wave = t >> 5;
    const int lane = t & 31;

    const float* W    = (wave == 0) ? Wq : (wave == 1) ? Wk : (wave == 2) ? Wv : Wsk;
    const float* bias = (wave == 0) ? bq : (wave == 1) ? bk : (wave == 2) ? bv : bsk;
    float*       out  = (wave == 0) ? q  : (wave == 1) ? k  : (wave == 2) ? v  : sk;

    const int m     = lane & 15;           // A row within tile / B,D column
    const int khalf = (lane >> 4) * 2;     // K sub-pair owned by this lane half

    v8f acc0 = {}, acc1 = {}, acc2 = {}, acc3 = {};

#pragma unroll
    for (int kb = 0; kb < 16; ++kb) {      // K = 64 in steps of 4
        const int kk = 4 * kb + khalf;
        v2f a;
        a.x = At[m * 68 + kk];
        a.y = At[m * 68 + kk + 1];

        v2f b0, b1, b2, b3;
        const float* Wr0 = W + (long)kk * HD_;        // row kk
        const float* Wr1 = W + (long)(kk + 1) * HD_;  // row kk+1
        b0.x = Wr0[ 0 + m]; b0.y = Wr1[ 0 + m];
        b1.x = Wr0[16 + m]; b1.y = Wr1[16 + m];
        b2.x = Wr0[32 + m]; b2.y = Wr1[32 + m];
        b3.x = Wr0[48 + m]; b3.y = Wr1[48 + m];

        acc0 = __builtin_amdgcn_wmma_f32_16x16x4_f32(false, a, false, b0, (short)0, acc0, false, false);
        acc1 = __builtin_amdgcn_wmma_f32_16x16x4_f32(false, a, false, b1, (short)0, acc1, false, false);
        acc2 = __builtin_amdgcn_wmma_f32_16x16x4_f32(false, a, false, b2, (short)0, acc2, false, false);
        acc3 = __builtin_amdgcn_wmma_f32_16x16x4_f32(false, a, false, b3, (short)0, acc3, false, false);
    }

    // D layout: VGPR r -> row (r + (lane>>4)*8), col (lane&15) of the tile.
    const int rbase = row0 + (lane >> 4) * 8;
#pragma unroll
    for (int nb = 0; nb < 4; ++nb) {
        const v8f acc = (nb == 0) ? acc0 : (nb == 1) ? acc1 : (nb == 2) ? acc2 : acc3;
        const float bvv = bias[nb * 16 + m];
#pragma unroll
        for (int r = 0; r < 8; ++r) {
            out[(long)(rbase + r) * HD_ + nb * 16 + m] = acc[r] + bvv;
        }
    }
}

// ---------------------------------------------------------------------------
// Kernel 2: CSR row offsets via binary search on the sorted edge_dst array.
// ---------------------------------------------------------------------------
__global__ void offsets_kernel(const int* __restrict__ dst, int* __restrict__ offs,
                               int n, int e)
{
    const int i = blockIdx.x * blockDim.x + threadIdx.x;
    if (i > n) return;
    int lo = 0, hi = e;
    while (lo < hi) {
        const int mid = (lo + hi) >> 1;
        lo = (dst[mid] < i) ? mid + 1 : lo;
        hi = (dst[mid] < i) ? hi      : mid;
    }
    offs[i] = lo;
}

// ---------------------------------------------------------------------------
// Kernel 3: per-destination-node wave32: online-softmax attention aggregation
// fused with gated skip, LayerNorm and PReLU. Lane owns channels 2*lane,
// 2*lane+1; head h = lane>>3 (group-of-8 shuffle reduce for per-head scores).
// Next edge's q/v rows are prefetched (global_prefetch_b8) to hide the
// dependent gather latency that dominates this bandwidth-bound phase.
// ---------------------------------------------------------------------------
__global__ __launch_bounds__(256) void attn_fused_kernel(
    const int* __restrict__ esrc, const int* __restrict__ offs,
    const float* __restrict__ q, const float* __restrict__ k,
    const float* __restrict__ v, const float* __restrict__ skip,
    const float* __restrict__ Wg, const float* __restrict__ bg,
    const float* __restrict__ ln_g, const float* __restrict__ ln_b,
    const float* __restrict__ prelu_a,
    float* __restrict__ out, int n)
{
    const int wave = threadIdx.x >> 5;
    const int node = blockIdx.x * 8 + wave;
    if (node >= n) return;
    const int lane = threadIdx.x & 31;
    const int c0 = lane * 2;
    const int c1 = c0 + 1;

    const long nb = (long)node * HD_;
    const float kd0 = k[nb + c0];
    const float kd1 = k[nb + c1];

    const int e0 = offs[node];
    const int e1 = offs[node + 1];

    float mrun = -INFINITY, denom = 0.f, acc0 = 0.f, acc1 = 0.f;
    for (int e = e0; e < e1; ++e) {
        const int s = esrc[e];
        const long sb = (long)s * HD_;
        if (e + 1 < e1) {
            const long pb = (long)esrc[e + 1] * HD_;
            __builtin_prefetch(q + pb + c0, 0, 3);
            __builtin_prefetch(v + pb + c0, 0, 3);
        }
        // coalesced 256B row gather: 32 lanes x 8B
        float p = q[sb + c0] * kd0 + q[sb + c1] * kd1;
        // per-head (group of 8 lanes) reduction
        p += __shfl_xor(p, 1);
        p += __shfl_xor(p, 2);
        p += __shfl_xor(p, 4);
        const float score = p * 0.25f;           // 1/sqrt(D), D=16

        const float mnew = fmaxf(mrun, score);
        const float corr = __expf(mrun - mnew);  // exp(-inf)=0 on first edge
        const float w    = __expf(score - mnew);
        denom = denom * corr + w;
        acc0  = acc0  * corr + w * v[sb + c0];
        acc1  = acc1  * corr + w * v[sb + c1];
        mrun  = mnew;
    }
    const float inv = (denom > 0.f) ? (1.f / denom) : 0.f;  // isolated node -> 0
    const float r0 = acc0 * inv;
    const float r1 = acc1 * inv;

    const float s0 = skip[nb + c0];
    const float s1 = skip[nb + c1];

    // gate = sigmoid([skip, rst, skip-rst] @ Wg + bg)
    float z = s0 * Wg[c0]       + r0 * Wg[64 + c0] + (s0 - r0) * Wg[128 + c0]
            + s1 * Wg[c1]       + r1 * Wg[64 + c1] + (s1 - r1) * Wg[128 + c1];
    z += __shfl_xor(z, 1);
    z += __shfl_xor(z, 2);
    z += __shfl_xor(z, 4);
    z += __shfl_xor(z, 8);
    z += __shfl_xor(z, 16);
    z += bg[0];
    const float g = 1.f / (1.f + __expf(-z));

    const float y0 = g * s0 + (1.f - g) * r0;
    const float y1 = g * s1 + (1.f - g) * r1;

    // LayerNorm over 64 channels (2 per lane)
    float sum = y0 + y1;
    sum += __shfl_xor(sum, 1);
    sum += __shfl_xor(sum, 2);
    sum += __shfl_xor(sum, 4);
    sum += __shfl_xor(sum, 8);
    sum += __shfl_xor(sum, 16);
    const float mu = sum * (1.f / 64.f);

    const float d0 = y0 - mu;
    const float d1 = y1 - mu;
    float vs = d0 * d0 + d1 * d1;
    vs += __shfl_xor(vs, 1);
    vs += __shfl_xor(vs, 2);
    vs += __shfl_xor(vs, 4);
    vs += __shfl_xor(vs, 8);
    vs += __shfl_xor(vs, 16);
    const float rstd = rsqrtf(vs * (1.f / 64.f) + 1e-5f);

    float o0 = d0 * rstd * ln_g[c0] + ln_b[c0];
    float o1 = d1 * rstd * ln_g[c1] + ln_b[c1];
    const float a = prelu_a[0];
    out[nb + c0] = (o0 >= 0.f) ? o0 : a * o0;
    out[nb + c1] = (o1 >= 0.f) ? o1 : a * o1;
}

// ---------------------------------------------------------------------------
extern "C" void kernel_launch(void* const* d_in, const int* in_sizes, int n_in,
                              void* d_out, int out_size, void* d_ws, size_t ws_size,
                              hipStream_t stream)
{
    const float* feat = (const float*)d_in[0];
    const int*   esrc = (const int*)d_in[1];
    const int*   edst = (const int*)d_in[2];
    const float* Wq  = (const float*)d_in[3];
    const float* bq  = (const float*)d_in[4];
    const float* Wk  = (const float*)d_in[5];
    const float* bk  = (const float*)d_in[6];
    const float* Wv  = (const float*)d_in[7];
    const float* bv  = (const float*)d_in[8];
    const float* Wsk = (const float*)d_in[9];
    const float* bsk = (const float*)d_in[10];
    const float* Wg  = (const float*)d_in[11];
    const float* bg  = (const float*)d_in[12];
    const float* lng = (const float*)d_in[13];
    const float* lnb = (const float*)d_in[14];
    const float* pa  = (const float*)d_in[15];
    float* out = (float*)d_out;

    // workspace layout
    float* q  = (float*)d_ws;
    float* k  = q  + (long)NNODES * HD_;
    float* v  = k  + (long)NNODES * HD_;
    float* sk = v  + (long)NNODES * HD_;
    int*  offs = (int*)(sk + (long)NNODES * HD_);

    // 1) projections (WMMA f32 16x16x4, TDM-staged A tiles)
    proj_wmma_kernel<<<NNODES / 16, 128, 0, stream>>>(
        feat, Wq, bq, Wk, bk, Wv, bv, Wsk, bsk, q, k, v, sk);

    // 2) CSR offsets from sorted edge_dst
    offsets_kernel<<<(NNODES + 1 + 255) / 256, 256, 0, stream>>>(
        edst, offs, NNODES, NEDGES);

    // 3) fused attention + gate + LN + PReLU: one wave32 per destination node
    attn_fused_kernel<<<(NNODES + 7) / 8, 256, 0, stream>>>(
        esrc, offs, q, k, v, sk, Wg, bg, lng, lnb, pa, out, NNODES);
}